// GCN_82575041232955
// MI455X (gfx1250) — compile-verified
//
#include <hip/hip_runtime.h>

typedef float v2f __attribute__((ext_vector_type(2)));
typedef float v8f __attribute__((ext_vector_type(8)));

static __device__ __forceinline__ float relu_(float v) { return v > 0.0f ? v : 0.0f; }

// ---------------- degree / normalization ----------------
__global__ void k_init_deg(float* deg, int n) {
  int i = blockIdx.x * blockDim.x + threadIdx.x;
  if (i < n) deg[i] = 1.0f;  // self-loop contributes 1 to every node's degree
}

__global__ void k_deg_scatter(const int* __restrict__ dst, float* deg, int e) {
  int i = blockIdx.x * blockDim.x + threadIdx.x;
  if (i < e) atomicAdd(&deg[dst[i]], 1.0f);
}

__global__ void k_deg_to_dinv(float* deg, int n) {
  int i = blockIdx.x * blockDim.x + threadIdx.x;
  if (i < n) deg[i] = rsqrtf(deg[i]);  // deg >= 1 always (self loops)
}

// ---------------- weight swizzle into WMMA B layout ----------------
// V_WMMA_F32_16X16X4_F32 B fragment (4x16, f32, 2 VGPRs):
//   lanes 0-15 : v0=B[k0+0][n], v1=B[k0+1][n]   (n = lane)
//   lanes 16-31: v0=B[k0+2][n], v1=B[k0+3][n]   (n = lane-16)
template <int NT>  // NT = number of 16-wide column tiles (C = 16*NT)
__global__ void k_swizzle_w(const float* __restrict__ W, float* __restrict__ Bsw) {
  int idx = blockIdx.x * blockDim.x + threadIdx.x;
  if (idx >= 32 * NT * 32) return;  // 32 k-steps * NT col tiles * 32 lanes
  int lane = idx & 31;
  int nt = (idx >> 5) % NT;
  int kk = idx / (32 * NT);
  const int C = NT * 16;
  int col = nt * 16 + (lane & 15);
  int krow = kk * 4 + 2 * (lane >> 4);
  Bsw[idx * 2 + 0] = W[(size_t)krow * C + col];
  Bsw[idx * 2 + 1] = W[(size_t)(krow + 1) * C + col];
}

// ---------------- fp32 WMMA GEMM: H[N x 16*NT] = X[N x 128] * W[128 x 16*NT] --------
// One wave per 16-row tile; NT v8f accumulators; K stepped 4 at a time (32 steps).
template <int NT>
__global__ __launch_bounds__(256) void k_gemm_wmma(const float* __restrict__ X,
                                                   const float* __restrict__ Bsw,
                                                   float* __restrict__ H,
                                                   int nRows, int nRowTiles) {
  const int lane = threadIdx.x & 31;
  const int wave = threadIdx.x >> 5;
  const int rt = blockIdx.x * 8 + wave;
  if (rt >= nRowTiles) return;  // wave-uniform: EXEC all-ones for WMMA below

  const int mrow = lane & 15;
  const int khalf = lane >> 4;  // 0: K=0,1  1: K=2,3
  int arow = rt * 16 + mrow;
  if (arow >= nRows) arow = nRows - 1;  // clamp loads; stores are guarded
  const float* aptr = X + (size_t)arow * 128 + 2 * khalf;

  v8f acc[NT];
#pragma unroll
  for (int t = 0; t < NT; ++t) acc[t] = (v8f)(0.0f);

#pragma unroll 4
  for (int kk = 0; kk < 32; ++kk) {
    v2f a = *(const v2f*)(aptr + kk * 4);
    const float* bptr = Bsw + ((size_t)kk * NT) * 64 + lane * 2;
#pragma unroll
    for (int t = 0; t < NT; ++t) {
      v2f b = *(const v2f*)(bptr + t * 64);
      acc[t] = __builtin_amdgcn_wmma_f32_16x16x4_f32(
          false, a, false, b, (short)0, acc[t], false, false);
    }
  }

  // C/D layout: VGPR r -> M = r + 8*khalf, N = lane&15
  const int orow0 = rt * 16 + 8 * khalf;
#pragma unroll
  for (int t = 0; t < NT; ++t) {
#pragma unroll
    for (int r = 0; r < 8; ++r) {
      int orow = orow0 + r;
      if (orow < nRows)
        H[(size_t)orow * (NT * 16) + t * 16 + mrow] = acc[t][r];
    }
  }
}

// ---------------- self-loop + bias init: O[i] = b + H[i] * dinv[i]^2 ----------------
template <int C>
__global__ void k_self_init(const float* __restrict__ H, const float* __restrict__ dinv,
                            const float* __restrict__ bias, float* __restrict__ out, int n) {
  const int C4 = C / 4;
  int idx = blockIdx.x * blockDim.x + threadIdx.x;
  if (idx >= n * C4) return;
  int i = idx / C4;
  int c4 = idx - i * C4;
  float di = dinv[i];
  float w = di * di;
  float4 h = ((const float4*)(H + (size_t)i * C))[c4];
  float4 b = ((const float4*)bias)[c4];
  float4 o;
  o.x = fmaf(h.x, w, b.x);
  o.y = fmaf(h.y, w, b.y);
  o.z = fmaf(h.z, w, b.z);
  o.w = fmaf(h.w, w, b.w);
  ((float4*)(out + (size_t)i * C))[c4] = o;
}

// ---------------- edge scatter: O[dst] += H[src] * dinv[src]*dinv[dst] --------------
// One wave per edge; C=128 -> float4/lane, C=64 -> float2/lane.
template <int C>
__global__ __launch_bounds__(256) void k_edge_scatter(const float* __restrict__ H,
                                                      const int* __restrict__ src,
                                                      const int* __restrict__ dst,
                                                      const float* __restrict__ dinv,
                                                      float* __restrict__ out, int nE) {
  int wid = (int)((blockIdx.x * (size_t)blockDim.x + threadIdx.x) >> 5);
  int lane = threadIdx.x & 31;
  if (wid >= nE) return;
  int s = src[wid];
  int d = dst[wid];
  float w = dinv[s] * dinv[d];
  if (C == 128) {
    float4 v = ((const float4*)(H + (size_t)s * 128))[lane];
    float* op = out + (size_t)d * 128 + lane * 4;
    atomicAdd(op + 0, v.x * w);
    atomicAdd(op + 1, v.y * w);
    atomicAdd(op + 2, v.z * w);
    atomicAdd(op + 3, v.w * w);
  } else {  // C == 64
    float2 v = ((const float2*)(H + (size_t)s * 64))[lane];
    float* op = out + (size_t)d * 64 + lane * 2;
    atomicAdd(op + 0, v.x * w);
    atomicAdd(op + 1, v.y * w);
  }
}

// ---------------- BatchNorm (C=128) ----------------
__global__ void k_zero_stats(float* stats) { stats[threadIdx.x] = 0.0f; }  // 256 threads

__global__ void k_bn_stats(const float* __restrict__ X, float* __restrict__ stats,
                           int n, int rowsPerBlock) {
  int c = threadIdx.x;  // blockDim = 128, one column per thread (coalesced rows)
  int r0 = blockIdx.x * rowsPerBlock;
  int r1 = r0 + rowsPerBlock;
  if (r1 > n) r1 = n;
  float s = 0.0f, s2 = 0.0f;
  for (int r = r0; r < r1; ++r) {
    float v = X[(size_t)r * 128 + c];
    s += v;
    s2 = fmaf(v, v, s2);
  }
  atomicAdd(&stats[c], s);
  atomicAdd(&stats[128 + c], s2);
}

__global__ void k_bn_finalize(const float* __restrict__ stats, const float* __restrict__ g,
                              const float* __restrict__ be, float* __restrict__ ss, float invN) {
  int c = threadIdx.x;  // blockDim = 128
  float mu = stats[c] * invN;
  float var = fmaf(-mu, mu, stats[128 + c] * invN);  // biased var, matches jnp.var
  float sc = g[c] * rsqrtf(var + 1e-5f);
  ss[c] = sc;
  ss[128 + c] = fmaf(-mu, sc, be[c]);
}

__global__ void k_bn_relu_inplace(float* __restrict__ X, const float* __restrict__ ss, int total4) {
  int idx = blockIdx.x * blockDim.x + threadIdx.x;
  if (idx >= total4) return;
  int c4 = idx & 31;  // 128/4 columns of float4
  float4 v = ((float4*)X)[idx];
  float4 sc = ((const float4*)ss)[c4];
  float4 sh = ((const float4*)(ss + 128))[c4];
  float4 o;
  o.x = relu_(fmaf(v.x, sc.x, sh.x));
  o.y = relu_(fmaf(v.y, sc.y, sh.y));
  o.z = relu_(fmaf(v.z, sc.z, sh.z));
  o.w = relu_(fmaf(v.w, sc.w, sh.w));
  ((float4*)X)[idx] = o;
}

// =====================================================================================
extern "C" void kernel_launch(void* const* d_in, const int* in_sizes, int n_in,
                              void* d_out, int out_size, void* d_ws, size_t ws_size,
                              hipStream_t stream) {
  const float* x = (const float*)d_in[0];
  const int* ei = (const int*)d_in[1];
  const float* W1 = (const float*)d_in[2];
  const float* b1 = (const float*)d_in[3];
  const float* g1 = (const float*)d_in[4];
  const float* be1 = (const float*)d_in[5];
  const float* W2 = (const float*)d_in[6];
  const float* b2 = (const float*)d_in[7];
  const float* g2 = (const float*)d_in[8];
  const float* be2 = (const float*)d_in[9];
  const float* W3 = (const float*)d_in[10];
  const float* b3 = (const float*)d_in[11];

  const int N = in_sizes[0] / 128;
  const int E = in_sizes[1] / 2;
  const int* srcI = ei;
  const int* dstI = ei + E;

  // ---- workspace partition ----
  char* p = (char*)d_ws;
  auto alloc = [&](size_t bytes) {
    char* r = p;
    p += (bytes + 255) & ~(size_t)255;
    return r;
  };
  float* dinv = (float*)alloc((size_t)N * 4);
  float* Hbuf = (float*)alloc((size_t)N * 128 * 4);  // GEMM output per layer
  float* Obuf = (float*)alloc((size_t)N * 128 * 4);  // aggregate / activation (in-place BN)
  float* Wsw1 = (float*)alloc(128 * 128 * 4);
  float* Wsw2 = (float*)alloc(128 * 128 * 4);
  float* Wsw3 = (float*)alloc(128 * 64 * 4);
  float* stats = (float*)alloc(256 * 4);
  float* ss = (float*)alloc(256 * 4);
  float* out = (float*)d_out;  // N x 64

  auto cdiv = [](long long a, long long b) { return (int)((a + b - 1) / b); };
  const int nRowTiles = cdiv(N, 16);
  const int gemmBlocks = cdiv(nRowTiles, 8);
  const int scatBlocks = cdiv((long long)E * 32, 256);
  const int bnRows = 256;
  const int bnBlocks = cdiv(N, bnRows);
  const float invN = 1.0f / (float)N;

  // ---- symmetric normalization coefficients ----
  k_init_deg<<<cdiv(N, 256), 256, 0, stream>>>(dinv, N);
  k_deg_scatter<<<cdiv(E, 256), 256, 0, stream>>>(dstI, dinv, E);
  k_deg_to_dinv<<<cdiv(N, 256), 256, 0, stream>>>(dinv, N);

  // ---- weight swizzles (tiny) ----
  k_swizzle_w<8><<<cdiv(32 * 8 * 32, 256), 256, 0, stream>>>(W1, Wsw1);
  k_swizzle_w<8><<<cdiv(32 * 8 * 32, 256), 256, 0, stream>>>(W2, Wsw2);
  k_swizzle_w<4><<<cdiv(32 * 4 * 32, 256), 256, 0, stream>>>(W3, Wsw3);

  // ================= Layer 1 =================
  k_gemm_wmma<8><<<gemmBlocks, 256, 0, stream>>>(x, Wsw1, Hbuf, N, nRowTiles);
  k_self_init<128><<<cdiv((long long)N * 32, 256), 256, 0, stream>>>(Hbuf, dinv, b1, Obuf, N);
  k_edge_scatter<128><<<scatBlocks, 256, 0, stream>>>(Hbuf, srcI, dstI, dinv, Obuf, E);
  k_zero_stats<<<1, 256, 0, stream>>>(stats);
  k_bn_stats<<<bnBlocks, 128, 0, stream>>>(Obuf, stats, N, bnRows);
  k_bn_finalize<<<1, 128, 0, stream>>>(stats, g1, be1, ss, invN);
  k_bn_relu_inplace<<<cdiv((long long)N * 32, 256), 256, 0, stream>>>(Obuf, ss, N * 32);

  // ================= Layer 2 =================
  k_gemm_wmma<8><<<gemmBlocks, 256, 0, stream>>>(Obuf, Wsw2, Hbuf, N, nRowTiles);
  k_self_init<128><<<cdiv((long long)N * 32, 256), 256, 0, stream>>>(Hbuf, dinv, b2, Obuf, N);
  k_edge_scatter<128><<<scatBlocks, 256, 0, stream>>>(Hbuf, srcI, dstI, dinv, Obuf, E);
  k_zero_stats<<<1, 256, 0, stream>>>(stats);
  k_bn_stats<<<bnBlocks, 128, 0, stream>>>(Obuf, stats, N, bnRows);
  k_bn_finalize<<<1, 128, 0, stream>>>(stats, g2, be2, ss, invN);
  k_bn_relu_inplace<<<cdiv((long long)N * 32, 256), 256, 0, stream>>>(Obuf, ss, N * 32);

  // ================= Layer 3 (output, C=64, no BN/ReLU) =================
  k_gemm_wmma<4><<<gemmBlocks, 256, 0, stream>>>(Obuf, Wsw3, Hbuf, N, nRowTiles);
  k_self_init<64><<<cdiv((long long)N * 16, 256), 256, 0, stream>>>(Hbuf, dinv, b3, out, N);
  k_edge_scatter<64><<<scatBlocks, 256, 0, stream>>>(Hbuf, srcI, dstI, dinv, out, E);
}